// Model_24850680774687
// MI455X (gfx1250) — compile-verified
//
#include <hip/hip_runtime.h>
#include <hip/hip_bf16.h>
#include <stdint.h>

// Problem constants (fixed by the reference).
#define B_      8
#define N_      262144
#define D_      64
#define S_      4096

// Decomposition.
#define DSLICE  16                         // dims per workgroup slice
#define NSLICES (D_ / DSLICE)              // 4
#define NCHUNK  16                         // N split across workgroups
#define ROWS_PER_CHUNK (N_ / NCHUNK)       // 16384
#define BLOCK   512                        // 16 waves (wave32)
#define WAVES   (BLOCK / 32)
#define ROWS_PER_WAVE_ITER  4              // 8 lanes/row * float2 = 16 floats/row
#define ROWS_PER_BLOCK_ITER (WAVES * ROWS_PER_WAVE_ITER)   // 64
#define ITERS   (ROWS_PER_CHUNK / ROWS_PER_BLOCK_ITER)     // 256
#define DEPTH   4                          // async pipeline depth (per wave)

#define ACC_ELEMS   (S_ * DSLICE)          // 65536 floats = 256 KB
#define STAGE_ELEMS (WAVES * DEPTH * 64)   // 4096 floats  =  16 KB
#define SMEM_BYTES  ((ACC_ELEMS + STAGE_ELEMS) * 4)        // 278528 <= 320 KB

// ---------------------------------------------------------------------------
// CDNA5 async global->LDS copy (ASYNCcnt path), GVS addressing, non-temporal:
//   each lane copies 8 bytes from global SADDR plus its VADDR offset to its LDS address in VDST
// X is streamed exactly once (512 MB) -> NT so it does not sweep the 192 MB
// L2 that we want to keep holding `keys` and the 8 MB output.
// ---------------------------------------------------------------------------
__device__ __forceinline__ void async_load_b64_to_lds_nt(unsigned lds_byte_off,
                                                         unsigned global_byte_off,
                                                         const void* sgpr_base) {
  asm volatile("global_load_async_to_lds_b64 %0, %1, %2 th:TH_LOAD_NT"
               :: "v"(lds_byte_off), "v"(global_byte_off), "s"(sgpr_base)
               : "memory");
}
__device__ __forceinline__ void wait_async_le3() {
  asm volatile("s_wait_asynccnt 0x3" ::: "memory");
}
__device__ __forceinline__ void wait_async_le0() {
  asm volatile("s_wait_asynccnt 0x0" ::: "memory");
}
// LDS ops complete in order: allowing 2 outstanding (the two ds_add_f32 that
// follow the stage ds_load_b64 in program order) guarantees the stage read
// has retired before the async engine overwrites that slot.
__device__ __forceinline__ void wait_ds_le2() {
  asm volatile("s_wait_dscnt 0x2" ::: "memory");
}

extern __shared__ float smem[];

__global__ void __launch_bounds__(BLOCK)
zero_f32_kernel(float* __restrict__ p, int n) {
  int i = blockIdx.x * BLOCK + threadIdx.x;
  if (i < n) p[i] = 0.0f;
}

__global__ void __launch_bounds__(BLOCK)
seg_sum_kernel(const float* __restrict__ X,
               const int*   __restrict__ keys,
               float*       __restrict__ out) {
  float* acc   = smem;               // [S_][DSLICE]        256 KB
  float* stage = smem + ACC_ELEMS;   // [WAVES][DEPTH][64]   16 KB

  const int bx    = blockIdx.x;
  const int chunk = bx % NCHUNK;
  const int slice = (bx / NCHUNK) % NSLICES;
  const int b     = bx / (NCHUNK * NSLICES);

  const int tid = threadIdx.x;

  // Zero the privatized full-segment accumulator (impossible in 64 KB CDNA4
  // LDS; CDNA5's 320 KB/WGP makes one-pass privatization work).
  for (int i = tid; i < ACC_ELEMS; i += BLOCK) acc[i] = 0.0f;
  __syncthreads();

  const int wave      = tid >> 5;
  const int lane      = tid & 31;
  const int rowInWave = lane >> 3;   // 0..3
  const int dpair     = lane & 7;    // 8 lanes * 2 floats = 16-dim slice

  const int rowBase = chunk * ROWS_PER_CHUNK + wave * ROWS_PER_WAVE_ITER + rowInWave;

  // Wave-uniform SGPR base; per-lane 32-bit byte offset (max ~67 MB, fits i32).
  const float* sbase = X + (size_t)b * N_ * D_ + slice * DSLICE;
  unsigned goff = (unsigned)rowBase * (D_ * 4) + (unsigned)dpair * 8u;
  const unsigned gstep = ROWS_PER_BLOCK_ITER * D_ * 4;   // 16384 B / block-iter

  // Per-lane LDS staging slots; low 32 bits of the flat LDS address are the
  // LDS byte offset per the gfx1250 aperture encoding.
  const float* stv[DEPTH];
  unsigned     ldsv[DEPTH];
#pragma unroll
  for (int u = 0; u < DEPTH; ++u) {
    stv[u]  = stage + (wave * DEPTH + u) * 64 + rowInWave * DSLICE + dpair * 2;
    ldsv[u] = (unsigned)(uintptr_t)stv[u];
  }

  const int* kptr = keys + rowBase;
  const int  accBase = dpair * 2;

  // Prologue: fill all DEPTH buffers (async load completions are in-order).
#pragma unroll
  for (int u = 0; u < DEPTH; ++u)
    async_load_b64_to_lds_nt(ldsv[u], goff + (unsigned)u * gstep, sbase);

  // Branch-free steady state, unrolled by DEPTH so buffer choice is static.
  int i = 0;
  for (; i + DEPTH < ITERS; i += DEPTH) {
#pragma unroll
    for (int u = 0; u < DEPTH; ++u) {
      const int idx = i + u;
      wait_async_le3();                                  // buffer idx landed
      const float2 v = *(const float2*)stv[u];
      const int    k = kptr[idx * ROWS_PER_BLOCK_ITER];
      __builtin_prefetch(kptr + (idx + 16) * ROWS_PER_BLOCK_ITER, 0, 0);
      unsafeAtomicAdd(&acc[k * DSLICE + accBase + 0], v.x);   // ds_add_f32
      unsafeAtomicAdd(&acc[k * DSLICE + accBase + 1], v.y);
      wait_ds_le2();                                     // stage read retired
      async_load_b64_to_lds_nt(ldsv[u], goff + (unsigned)(u + DEPTH) * gstep, sbase);
    }
    goff += DEPTH * gstep;
  }
  // Epilogue: drain the last DEPTH iterations, no refills.
#pragma unroll
  for (int u = 0; u < DEPTH; ++u, ++i) {
    wait_async_le0();
    const float2 v = *(const float2*)stv[i & (DEPTH - 1)];
    const int    k = kptr[i * ROWS_PER_BLOCK_ITER];
    unsafeAtomicAdd(&acc[k * DSLICE + accBase + 0], v.x);
    unsafeAtomicAdd(&acc[k * DSLICE + accBase + 1], v.y);
  }
  __syncthreads();

  // Flush privatized accumulator with native f32 global atomics (L2-resident).
  const size_t outBase = (size_t)b * S_ * D_ + slice * DSLICE;
  for (int j = tid; j < ACC_ELEMS; j += BLOCK) {
    const int s = j >> 4;            // segment
    const int d = j & (DSLICE - 1);  // dim within slice
    unsafeAtomicAdd(&out[outBase + (size_t)s * D_ + d], acc[j]);
  }
}

extern "C" void kernel_launch(void* const* d_in, const int* in_sizes, int n_in,
                              void* d_out, int out_size, void* d_ws, size_t ws_size,
                              hipStream_t stream) {
  const float* X    = (const float*)d_in[0];
  const int*   keys = (const int*)d_in[1];
  float*       out  = (float*)d_out;

  (void)n_in; (void)d_ws; (void)ws_size; (void)in_sizes;

  static_assert(SMEM_BYTES <= 320 * 1024, "LDS budget");
  hipFuncSetAttribute((const void*)seg_sum_kernel,
                      hipFuncAttributeMaxDynamicSharedMemorySize, SMEM_BYTES);

  // d_out is poisoned by the harness: zero it every call, then scatter-add.
  zero_f32_kernel<<<(out_size + BLOCK - 1) / BLOCK, BLOCK, 0, stream>>>(out, out_size);

  seg_sum_kernel<<<dim3(B_ * NSLICES * NCHUNK), dim3(BLOCK), SMEM_BYTES, stream>>>(
      X, keys, out);
}